// ColorExtractor_30236569764047
// MI455X (gfx1250) — compile-verified
//
#include <hip/hip_runtime.h>

// K-means color extractor for MI455X (gfx1250, wave32).
// One workgroup per image; centroids in LDS; distance dot-products on the
// matrix pipe via V_WMMA_F32_16X16X4_F32 (C=3 padded to 4).
//
// Mapping per 16-pixel tile:
//   A (16x4 f32)  = centroids x channels   (invariant across tiles)
//   B (4x16 f32)  = channels  x pixels     (tile data)
//   D (16x16 f32) = cluster-pixel dot products
// D layout: lane p (0-15) holds pixel p vs clusters 0-7 in acc[0..7];
//           lane p+16 holds pixel p vs clusters 8-15. One shfl_xor(16)
//           merges the two candidate sets; strict '<' keeps the lower
//           cluster index on ties for lanes 0-15 (the lanes that accumulate),
//           matching argmin tie-breaking.

typedef __attribute__((ext_vector_type(2))) float v2f;
typedef __attribute__((ext_vector_type(8))) float v8f;

#define B_IMGS  64
#define NPIX    (224 * 224)     // 50176
#define KC      16
#define CH      3
#define ITERS   10
#define TPB     1024
#define NWAVES  (TPB / 32)      // 32
#define NTILES  (NPIX / 16)     // 3136, divisible by NWAVES -> no divergence

__global__ __launch_bounds__(TPB)
void ColorExtractor_kmeans_kernel(const float* __restrict__ x_all,
                                  float* __restrict__ out) {
    __shared__ float s_cent[KC][CH];   // current centroids
    __shared__ float s_sums[KC][CH];   // per-cluster channel sums
    __shared__ float s_cnt[KC];        // per-cluster counts

    const int img = blockIdx.x;
    const float* __restrict__ x = x_all + (size_t)img * NPIX * CH;

    const int tid    = threadIdx.x;
    const int lane   = tid & 31;
    const int wave   = tid >> 5;
    const int lane16 = lane & 15;
    const bool hi    = (lane >= 16);

    // ---- deterministic centroid init: 16 hash-spread pixels ----
    if (tid < KC) {
        unsigned p = ((unsigned)(tid + 1) * 2654435761u) % (unsigned)NPIX;
        const float* xp = x + (size_t)p * CH;
        s_cent[tid][0] = xp[0];
        s_cent[tid][1] = xp[1];
        s_cent[tid][2] = xp[2];
        s_sums[tid][0] = 0.f; s_sums[tid][1] = 0.f; s_sums[tid][2] = 0.f;
        s_cnt[tid] = 0.f;
    }
    __syncthreads();

    for (int it = 0; it < ITERS; ++it) {
        // A fragment: 16x4 f32 centroid matrix (channel 3 zero-padded).
        // ISA layout: lanes 0-15 -> {K=0,K=1}, lanes 16-31 -> {K=2,K=3}.
        v2f a;
        a.x = s_cent[lane16][hi ? 2 : 0];
        a.y = hi ? 0.f : s_cent[lane16][1];

        // |c|^2 for this lane's 8 clusters (v + 8*hi).
        const int kbase = hi ? 8 : 0;
        float cc[8];
        #pragma unroll
        for (int v = 0; v < 8; ++v) {
            const float c0 = s_cent[v + kbase][0];
            const float c1 = s_cent[v + kbase][1];
            const float c2 = s_cent[v + kbase][2];
            cc[v] = c0 * c0 + c1 * c1 + c2 * c2;
        }

        // ---- assignment + accumulation sweep over 16-pixel tiles ----
        for (int t = wave; t < NTILES; t += NWAVES) {
            const int p = t * 16 + lane16;
            const float* xp = x + (size_t)p * CH;

            if (t + NWAVES < NTILES) {  // wave-uniform branch
                __builtin_prefetch(x + (size_t)(t + NWAVES) * 16 * CH + lane * 6, 0, 0);
            }

            // B fragment: 4x16 f32 pixel tile (column = pixel lane16).
            // lanes 0-15 -> {ch0, ch1}, lanes 16-31 -> {ch2, 0}.
            v2f b;
            b.x = xp[hi ? 2 : 0];
            const float x1 = xp[1];
            b.y = hi ? 0.f : x1;

            v8f acc = {};
            // D[k][p] = sum_c cent[k][c] * x[p][c]
            acc = __builtin_amdgcn_wmma_f32_16x16x4_f32(
                /*neg_a=*/false, a, /*neg_b=*/false, b,
                /*c_mod=*/(short)0, acc, /*reuse_a=*/false, /*reuse_b=*/false);

            // Local argmin over this lane's 8 clusters (branchless selects).
            // score = |c|^2 - 2*dot  (|x|^2 constant per pixel, irrelevant)
            float best  = __builtin_fmaf(-2.f, acc[0], cc[0]);
            int   bestk = kbase;
            #pragma unroll
            for (int v = 1; v < 8; ++v) {
                const float s = __builtin_fmaf(-2.f, acc[v], cc[v]);
                const bool lt = (s < best);
                best  = lt ? s : best;
                bestk = lt ? (v + kbase) : bestk;
            }
            // Branchless merge with partner half-wave. Strict '<' => on tie,
            // lanes 0-15 keep their (lower-index) cluster == argmin semantics.
            const float obest = __shfl_xor(best, 16, 32);
            const int   obk   = __shfl_xor(bestk, 16, 32);
            const bool  take  = (obest < best);
            bestk = take ? obk : bestk;

            // Lanes 0-15 each own one pixel: accumulate into cluster bins.
            if (!hi) {
                atomicAdd(&s_sums[bestk][0], xp[0]);
                atomicAdd(&s_sums[bestk][1], x1);
                atomicAdd(&s_sums[bestk][2], xp[2]);
                atomicAdd(&s_cnt[bestk], 1.f);
            }
        }
        __syncthreads();

        // ---- centroid update (matches ref: sums / counts, no guard) ----
        if (tid < KC) {
            const float inv = 1.f / s_cnt[tid];
            s_cent[tid][0] = s_sums[tid][0] * inv;
            s_cent[tid][1] = s_sums[tid][1] * inv;
            s_cent[tid][2] = s_sums[tid][2] * inv;
            s_sums[tid][0] = 0.f; s_sums[tid][1] = 0.f; s_sums[tid][2] = 0.f;
            s_cnt[tid] = 0.f;
        }
        __syncthreads();
    }

    // Output: B x (K*C), row-major per image.
    if (tid < KC) {
        float* o = out + (size_t)img * (KC * CH) + tid * CH;
        o[0] = s_cent[tid][0];
        o[1] = s_cent[tid][1];
        o[2] = s_cent[tid][2];
    }
}

extern "C" void kernel_launch(void* const* d_in, const int* in_sizes, int n_in,
                              void* d_out, int out_size, void* d_ws, size_t ws_size,
                              hipStream_t stream) {
    (void)in_sizes; (void)n_in; (void)d_ws; (void)ws_size; (void)out_size;
    const float* x = (const float*)d_in[0];
    float* out = (float*)d_out;
    ColorExtractor_kmeans_kernel<<<B_IMGS, TPB, 0, stream>>>(x, out);
}